// Agent_44873818308719
// MI455X (gfx1250) — compile-verified
//
#include <hip/hip_runtime.h>

typedef float v2f __attribute__((ext_vector_type(2)));
typedef float v4f __attribute__((ext_vector_type(4)));
typedef float v8f __attribute__((ext_vector_type(8)));

#define DIMS 128        // feature dims
#define MPROTO 32       // prototypes per batch
#define LDS_STRIDE 130  // 128 + 2 pad floats: kills LDS bank conflicts on B reads

// One wave -> one 16-row query tile x all 32 prototypes (two 16x16 WMMA D tiles).
// D = Q x P^T accumulated f32 via V_WMMA_F32_16X16X4_F32.
// ||p||^2 precomputed once per block into LDS; ||q||^2 fused into the A-fragment loads.
__global__ __launch_bounds__(128) void Agent_proto_sqdist_kernel(
    const float* __restrict__ protos,   // (bs, 32, 128)
    const float* __restrict__ query,    // (bs, n, 128)
    const float* __restrict__ scale,    // (1,)
    float* __restrict__ out,            // (bs, n, 32)
    int n, int tiles_per_batch) {
  __shared__ float sm[MPROTO * LDS_STRIDE];
  __shared__ float smpp[MPROTO];

  const int tid  = threadIdx.x;
  const int wave = tid >> 5;
  const int lane = tid & 31;
  const int h    = lane >> 4;   // half-wave selector: picks K sub-pair per ISA A/B layout
  const int l16  = lane & 15;

  const int t     = blockIdx.x * 4 + wave;          // global 16-row tile id
  const int batch = t / tiles_per_batch;            // all 4 waves share a batch (256 % 4 == 0)
  const int n0    = (t % tiles_per_batch) * 16;

  const float* pbase = protos + (size_t)batch * MPROTO * DIMS;

  // ---- Stage this batch's prototypes (16 KB) into LDS, padded stride ----
  #pragma unroll
  for (int j = 0; j < (MPROTO * DIMS / 2) / 128; ++j) {   // 2048 float2 / 128 threads
    int idx = j * 128 + tid;
    int row = idx >> 6;        // 64 float2 per row
    int c2  = (idx & 63) * 2;
    v2f v = *(const v2f*)(pbase + row * DIMS + c2);
    *(v2f*)(&sm[row * LDS_STRIDE + c2]) = v;       // (row*130 + c2)*4 bytes, 8B aligned
  }

  // ---- ||p||^2 once per block: threads 0..31 each own one prototype row ----
  if (tid < MPROTO) {
    const float* pr = pbase + tid * DIMS;
    float acc = 0.f;
    #pragma unroll
    for (int k = 0; k < DIMS; k += 4) {
      v4f p = *(const v4f*)(pr + k);                // stays hot in L0 from staging
      acc += p.x * p.x + p.y * p.y + p.z * p.z + p.w * p.w;
    }
    smpp[tid] = acc;
  }
  __syncthreads();

  // ---- K loop: 32 x (1 A-frag global load, 2 B-frag LDS loads, 1 fma, 2 WMMA) ----
  const float* qrow = query + ((size_t)batch * n + n0 + l16) * DIMS + 2 * h;
  const float* b0p  = &sm[l16 * LDS_STRIDE + 2 * h];
  const float* b1p  = &sm[(16 + l16) * LDS_STRIDE + 2 * h];

  v8f c0 = {};
  v8f c1 = {};
  float qpart = 0.f;

  #pragma unroll
  for (int k = 0; k < DIMS; k += 4) {
    v2f a  = *(const v2f*)(qrow + k);   // A[M=l16][k+2h : k+2h+1]
    v2f b0 = *(const v2f*)(b0p + k);    // B = P^T, cols 0..15
    v2f b1 = *(const v2f*)(b1p + k);    // B = P^T, cols 16..31
    qpart += a.x * a.x + a.y * a.y;     // free ||q||^2 partial
    // (neg_a, A, neg_b, B, c_mod, C, reuse_a, reuse_b)
    c0 = __builtin_amdgcn_wmma_f32_16x16x4_f32(false, a, false, b0, (short)0, c0, false, false);
    c1 = __builtin_amdgcn_wmma_f32_16x16x4_f32(false, a, false, b1, (short)0, c1, false, false);
  }

  // Lane L and L^16 cover disjoint K halves of the same query row -> one xor-fold.
  float qq  = qpart + __shfl_xor(qpart, 16, 32);    // lane L holds ||q_{n0+l16}||^2
  float pp0 = smpp[l16];                            // col L%16 matches D layout directly
  float pp1 = smpp[16 + l16];

  const float s = scale[0] * (1.0f / DIMS);
  float* obase = out + (size_t)batch * n * MPROTO;

  // D layout: VGPR r, lane L -> row r + 8*(L/16), col L%16.
  #pragma unroll
  for (int r = 0; r < 8; ++r) {
    int row = r + 8 * h;
    float qq_r = __shfl(qq, row, 32);   // bpermute: fetch ||q_row||^2 into this lane
    size_t off = (size_t)(n0 + row) * MPROTO + l16;
    obase[off]      = (qq_r + pp0 - 2.0f * c0[r]) * s;
    obase[off + 16] = (qq_r + pp1 - 2.0f * c1[r]) * s;
  }
}

extern "C" void kernel_launch(void* const* d_in, const int* in_sizes, int n_in,
                              void* d_out, int out_size, void* d_ws, size_t ws_size,
                              hipStream_t stream) {
  (void)n_in; (void)out_size; (void)d_ws; (void)ws_size;
  const float* protos = (const float*)d_in[0];  // prototypes (bs, 32, 128)
  const float* query  = (const float*)d_in[2];  // query      (bs, n, 128)
  const float* scale  = (const float*)d_in[7];  // scale      (1,)
  float* out = (float*)d_out;

  const int bs = in_sizes[0] / (MPROTO * DIMS); // 8
  const int n  = in_sizes[2] / (bs * DIMS);     // 4096
  const int tiles_per_batch = n / 16;           // 256
  const int blocks = (bs * tiles_per_batch) / 4;

  Agent_proto_sqdist_kernel<<<blocks, 128, 0, stream>>>(protos, query, scale, out,
                                                        n, tiles_per_batch);
}